// DistillLossContrastive_28896539967630
// MI455X (gfx1250) — compile-verified
//
#include <hip/hip_runtime.h>
#include <cfloat>

// Problem constants (match reference)
#define BS      8
#define NPTS    8192
#define NM      32
#define D       512
#define NTOT    (BS * NM)      // 256 masks total
#define KC      256            // K-chunk per workgroup (LDS tile = 32*256*4 = 32KB)
#define KCHUNKS (NPTS / KC)    // 32

typedef __attribute__((ext_vector_type(2))) float v2f;
typedef __attribute__((ext_vector_type(8))) float v8f;

// ---------------------------------------------------------------------------
// Kernel 0: zero the sum_feats accumulator (workspace is poisoned by harness)
// ---------------------------------------------------------------------------
__global__ void zero_kernel(float* __restrict__ p, int n) {
    int i = blockIdx.x * blockDim.x + threadIdx.x;
    if (i < n) p[i] = 0.0f;
}

// ---------------------------------------------------------------------------
// Kernel 1: per-mask visible-point counts  cnt[j] = sum_p mask_pts[j][p]
// ---------------------------------------------------------------------------
__global__ __launch_bounds__(256) void count_kernel(const float* __restrict__ mask_pts,
                                                    float* __restrict__ cnt) {
    const int j = blockIdx.x;                       // 0..255 (b*NM + m)
    const float* row = mask_pts + (size_t)j * NPTS;
    float s = 0.0f;
    for (int p = threadIdx.x; p < NPTS; p += 256) s += row[p];
    __shared__ float red[256];
    red[threadIdx.x] = s;
    __syncthreads();
    for (int off = 128; off > 0; off >>= 1) {
        if (threadIdx.x < off) red[threadIdx.x] += red[threadIdx.x + off];
        __syncthreads();
    }
    if (threadIdx.x == 0) cnt[j] = red[0];
}

// ---------------------------------------------------------------------------
// Kernel 2: WMMA batched GEMM  sum_feats[b*NM+m][d] += sum_{k-chunk} A*B
//   A = mask_pts[b]  (M=32 x K=8192), staged per-chunk in LDS
//   B = net_out[b]   (K=8192 x N=512), streamed from HBM exactly once
// Grid: (BS, KCHUNKS), Block: 512 threads = 16 waves.
// Each wave: two 16-wide N-subtiles, both 16-row M-tiles per subtile.
// ---------------------------------------------------------------------------
__global__ __launch_bounds__(512) void wmma_bmm_kernel(const float* __restrict__ net_out,
                                                       const float* __restrict__ mask_pts,
                                                       float* __restrict__ sum_feats) {
    const int b  = blockIdx.x;         // object
    const int kc = blockIdx.y;         // K-chunk
    const int k0 = kc * KC;

    __shared__ float ldsA[NM * KC];    // 32 x 256 f32 = 32KB

    // cooperative, coalesced load of the mask_pts chunk
    const float* mp = mask_pts + (size_t)b * NM * NPTS + k0;
    for (int idx = threadIdx.x; idx < NM * KC; idx += 512) {
        const int m = idx >> 8;        // / KC
        const int p = idx & (KC - 1);
        ldsA[idx] = mp[(size_t)m * NPTS + p];
    }
    __syncthreads();

    const int wave  = threadIdx.x >> 5;   // 0..15
    const int lane  = threadIdx.x & 31;
    const int l16   = lane & 15;
    const int lhalf = lane >> 4;          // 0 or 1 -> selects K pair {0,1} vs {2,3}

    const float* net = net_out + (size_t)b * NPTS * D + (size_t)k0 * D;

    for (int s = 0; s < 2; ++s) {
        const int n0 = wave * 32 + s * 16;    // N-subtile start (0..496)
        v8f acc0 = {};                        // rows m = 0..15
        v8f acc1 = {};                        // rows m = 16..31

        for (int k = 0; k < KC; k += 4) {
            // B fragment (4x16 f32): VGPR0 -> K = k+2*lhalf, VGPR1 -> K = k+2*lhalf+1
            const float* bp = net + (size_t)(k + 2 * lhalf) * D + n0 + l16;
            v2f bf;
            bf.x = bp[0];
            bf.y = bp[D];

            // A fragments (16x4 f32): lane row = l16 (+16 for second M-tile)
            const float* ap0 = &ldsA[(size_t)l16 * KC + k + 2 * lhalf];
            v2f a0; a0.x = ap0[0]; a0.y = ap0[1];
            const float* ap1 = &ldsA[(size_t)(16 + l16) * KC + k + 2 * lhalf];
            v2f a1; a1.x = ap1[0]; a1.y = ap1[1];

            acc0 = __builtin_amdgcn_wmma_f32_16x16x4_f32(
                false, a0, false, bf, (short)0, acc0, false, false);
            acc1 = __builtin_amdgcn_wmma_f32_16x16x4_f32(
                false, a1, false, bf, (short)0, acc1, false, false);
        }

        // C/D layout: VGPR r holds (M=r, N=lane) for lanes 0-15, (M=r+8) for 16-31
        const int col = n0 + l16;
        #pragma unroll
        for (int r = 0; r < 8; ++r) {
            const int row = r + 8 * lhalf;
            atomicAdd(&sum_feats[((size_t)(b * NM + row)) * D + col],      acc0[r]);
            atomicAdd(&sum_feats[((size_t)(b * NM + 16 + row)) * D + col], acc1[r]);
        }
    }
}

// ---------------------------------------------------------------------------
// Kernel 3: logits[i][t] = (mask_embs[i] . sum_feats[t]/(cnt[t]+1e-12)) * e^ls
// ---------------------------------------------------------------------------
__global__ __launch_bounds__(256) void logits_kernel(const float* __restrict__ mask_embs,
                                                     const float* __restrict__ sum_feats,
                                                     const float* __restrict__ cnt,
                                                     const float* __restrict__ logit_scale,
                                                     float* __restrict__ logits) {
    __shared__ float emb[D];
    const int i = blockIdx.x;
    for (int dd = threadIdx.x; dd < D; dd += 256)
        emb[dd] = mask_embs[(size_t)i * D + dd];
    __syncthreads();

    const int t = threadIdx.x;
    const float inv   = 1.0f / (cnt[t] + 1e-12f);
    const float scale = expf(logit_scale[0]);
    const float* sf = sum_feats + (size_t)t * D;
    float acc = 0.0f;
    for (int dd = 0; dd < D; ++dd) acc = fmaf(emb[dd], sf[dd], acc);
    logits[(size_t)i * NTOT + t] = acc * inv * scale;
}

// ---------------------------------------------------------------------------
// Kernel 4: symmetric InfoNCE diag losses + nonzero averages -> scalar
// ---------------------------------------------------------------------------
__global__ __launch_bounds__(256) void loss_kernel(const float* __restrict__ logits,
                                                   const float* __restrict__ cnt,
                                                   float* __restrict__ out) {
    const int i = threadIdx.x;
    const bool valid = (cnt[i] != 0.0f);
    const float diag = logits[(size_t)i * NTOT + i];

    // row i: texts_loss
    float rmax = -FLT_MAX;
    for (int j = 0; j < NTOT; ++j) rmax = fmaxf(rmax, logits[(size_t)i * NTOT + j]);
    float rsum = 0.0f;
    for (int j = 0; j < NTOT; ++j) rsum += expf(logits[(size_t)i * NTOT + j] - rmax);
    const float tl = valid ? (logf(rsum) + rmax - diag) : 0.0f;

    // column i: pts_loss (softmax over logits.T rows)
    float cmax = -FLT_MAX;
    for (int j = 0; j < NTOT; ++j) cmax = fmaxf(cmax, logits[(size_t)j * NTOT + i]);
    float csum = 0.0f;
    for (int j = 0; j < NTOT; ++j) csum += expf(logits[(size_t)j * NTOT + i] - cmax);
    const float pl = valid ? (logf(csum) + cmax - diag) : 0.0f;

    __shared__ float st[256], sp[256], ct[256], cp[256];
    st[i] = tl;
    sp[i] = pl;
    ct[i] = (tl > 0.0f) ? 1.0f : 0.0f;
    cp[i] = (pl > 0.0f) ? 1.0f : 0.0f;
    __syncthreads();
    for (int off = 128; off > 0; off >>= 1) {
        if (i < off) {
            st[i] += st[i + off];
            sp[i] += sp[i + off];
            ct[i] += ct[i + off];
            cp[i] += cp[i + off];
        }
        __syncthreads();
    }
    if (i == 0) {
        const float at = (ct[0] > 0.0f) ? st[0] / fmaxf(ct[0], 1.0f) : 0.0f;
        const float ap = (cp[0] > 0.0f) ? sp[0] / fmaxf(cp[0], 1.0f) : 0.0f;
        out[0] = 0.5f * (at + ap);
    }
}

// ---------------------------------------------------------------------------
// Host launcher
// ---------------------------------------------------------------------------
extern "C" void kernel_launch(void* const* d_in, const int* in_sizes, int n_in,
                              void* d_out, int out_size, void* d_ws, size_t ws_size,
                              hipStream_t stream) {
    const float* net_out     = (const float*)d_in[0];  // [BS*NPTS, D]
    const float* mask_embs   = (const float*)d_in[1];  // [NTOT, D]
    const float* mask_pts    = (const float*)d_in[2];  // [BS, NM, NPTS]
    const float* logit_scale = (const float*)d_in[3];  // scalar
    // d_in[4] = pt_offset: uniform per-object layout, not needed

    float* ws        = (float*)d_ws;
    float* sum_feats = ws;                 // NTOT*D      = 131072 floats
    float* cnt       = ws + NTOT * D;      // NTOT        = 256 floats
    float* logits    = cnt + NTOT;         // NTOT*NTOT   = 65536 floats

    const int nsf = NTOT * D;
    zero_kernel<<<(nsf + 255) / 256, 256, 0, stream>>>(sum_feats, nsf);
    count_kernel<<<NTOT, 256, 0, stream>>>(mask_pts, cnt);
    wmma_bmm_kernel<<<dim3(BS, KCHUNKS), 512, 0, stream>>>(net_out, mask_pts, sum_feats);
    logits_kernel<<<NTOT, 256, 0, stream>>>(mask_embs, sum_feats, cnt, logit_scale, logits);
    loss_kernel<<<1, 256, 0, stream>>>(logits, cnt, (float*)d_out);
}